// ReRanker_30975304139056
// MI455X (gfx1250) — compile-verified
//
#include <hip/hip_runtime.h>

typedef __attribute__((ext_vector_type(2))) float v2f;
typedef __attribute__((ext_vector_type(8))) float v8f;

#define EPS     1e-8f
#define BATCH   256
#define CNUM    128
#define DNUM    1024
#define BPAD    16                      // LDS row pad: 4 rows land on distinct banks

__launch_bounds__(256, 1)
__global__ void rerank_wmma_kernel(const float* __restrict__ doc,
                                   const float* __restrict__ summ,
                                   const float* __restrict__ cand,
                                   float* __restrict__ out)
{
    const int b    = blockIdx.x;
    const int tid  = threadIdx.x;
    const int lane = tid & 31;
    const int w    = tid >> 5;          // wave id, 0..7

    // B-matrix columns for the fused WMMA GEMM, stored as rows:
    // row 0 = doc[b], row 1 = summary[b], row 2 = s[b], row 3 = zeros
    __shared__ float bmat[4][DNUM + BPAD];
    __shared__ float rcn_s[CNUM];       // 1 / ||cand[b,c]||
    __shared__ float dotd_s[CNUM];      // cand·doc
    __shared__ float dots_s[CNUM];      // cand·summary
    __shared__ float doti_s[CNUM];      // cand·s
    __shared__ float red[3][8];
    __shared__ float scal[3];           // 1/dn, 1/sn, doc·summary

    const float* Db = doc  + (size_t)b * DNUM;
    const float* Sb = summ + (size_t)b * DNUM;
    const float* Cb = cand + (size_t)b * CNUM * DNUM;

    // ---------------- Phase 0: stage doc/summary in LDS; norms + doc·summary
    {
        float4 dv = *(const float4*)(Db + tid * 4);
        float4 sv = *(const float4*)(Sb + tid * 4);
        *(float4*)(&bmat[0][tid * 4]) = dv;
        *(float4*)(&bmat[1][tid * 4]) = sv;
        *(float4*)(&bmat[3][tid * 4]) = make_float4(0.f, 0.f, 0.f, 0.f);
        float dd = dv.x*dv.x + dv.y*dv.y + dv.z*dv.z + dv.w*dv.w;
        float ss = sv.x*sv.x + sv.y*sv.y + sv.z*sv.z + sv.w*sv.w;
        float ds = dv.x*sv.x + dv.y*sv.y + dv.z*sv.z + dv.w*sv.w;
        #pragma unroll
        for (int off = 16; off > 0; off >>= 1) {
            dd += __shfl_xor(dd, off, 32);
            ss += __shfl_xor(ss, off, 32);
            ds += __shfl_xor(ds, off, 32);
        }
        if (lane == 0) { red[0][w] = dd; red[1][w] = ss; red[2][w] = ds; }
    }
    __syncthreads();
    if (tid == 0) {
        float dd = 0.f, ss = 0.f, ds = 0.f;
        #pragma unroll
        for (int i = 0; i < 8; ++i) { dd += red[0][i]; ss += red[1][i]; ds += red[2][i]; }
        scal[0] = 1.0f / fmaxf(sqrtf(dd), EPS);
        scal[1] = 1.0f / fmaxf(sqrtf(ss), EPS);
        scal[2] = ds;
    }
    __syncthreads();

    // ---------------- Phase 1: per-candidate-row norms (HBM pass over 512 KB)
    #pragma unroll 1
    for (int i = 0; i < 16; ++i) {
        const int row = w * 16 + i;
        const float* rp = Cb + (size_t)row * DNUM;
        float acc = 0.f;
        #pragma unroll
        for (int j = 0; j < 8; ++j) {
            float4 v = *(const float4*)(rp + (lane + j * 32) * 4);
            acc += v.x*v.x + v.y*v.y + v.z*v.z + v.w*v.w;
        }
        #pragma unroll
        for (int off = 16; off > 0; off >>= 1) acc += __shfl_xor(acc, off, 32);
        if (lane == 0) rcn_s[row] = 1.0f / fmaxf(sqrtf(acc), EPS);
    }
    __syncthreads();

    // ---------------- Phase 2: s[d] = sum_c cand[c,d] / cn[c]  (L2 pass)
    {
        float4 acc = make_float4(0.f, 0.f, 0.f, 0.f);
        const float* cp = Cb + tid * 4;
        #pragma unroll 4
        for (int c = 0; c < CNUM; ++c) {
            float4 v = *(const float4*)(cp + (size_t)c * DNUM);
            float r = rcn_s[c];
            acc.x += v.x * r; acc.y += v.y * r;
            acc.z += v.z * r; acc.w += v.w * r;
        }
        *(float4*)(&bmat[2][tid * 4]) = acc;
    }
    __syncthreads();

    // ---------------- Phase 3: fused GEMM via full-fp32 WMMA  (L2 pass)
    // D[128x16] = cand[128x1024] x B[1024x16], B cols: 0=doc, 1=summary, 2=s.
    // Each wave owns a 16-row M tile; 256 x v_wmma_f32_16x16x4_f32 per wave.
    // Per-lane B base pointer picked ONCE (no selects / divergence in the loop).
    {
        const int m     = lane & 15;            // M row within tile == N column
        const int khalf = (lane >> 4) << 1;     // K sub-offset: 0 or 2
        const int brow  = (m < 3) ? m : 3;      // lanes 3..15 read the zero row
        const float* ap = Cb + (size_t)(w * 16 + m) * DNUM + khalf;
        const float* bp = &bmat[brow][khalf];

        v8f acc = {};
        #pragma unroll 8
        for (int k = 0; k < DNUM; k += 4) {
            // A: lanes 0-15 hold K={k,k+1}, lanes 16-31 hold K={k+2,k+3}
            v2f a  = *(const v2f*)(ap + k);
            // B: same K split; row choice baked into bp
            v2f bb = *(const v2f*)(bp + k);
            acc = __builtin_amdgcn_wmma_f32_16x16x4_f32(
                false, a, false, bb, (short)0, acc, false, false);
        }

        // D layout: VGPR r -> M = r (lanes 0-15) / r+8 (lanes 16-31), N = lane&15
        if (m < 3) {
            float* outp = (m == 0) ? dotd_s : (m == 1) ? dots_s : doti_s;
            const int rbase = w * 16 + ((lane >> 4) << 3);
            #pragma unroll
            for (int r = 0; r < 8; ++r)
                outp[rbase + r] = acc[r];
        }
    }
    __syncthreads();

    // ---------------- Phase 4: final scores
    const float rdn = scal[0];
    const float rsn = scal[1];

    if (tid < CNUM) {
        const int c = tid;
        const float rc = rcn_s[c];
        // outer_score[b,c] = (cand·doc) / (cn * dn)
        out[(size_t)b * CNUM + c] = dotd_s[c] * rc * rdn;
        // inner_score[b,c] = (chat·s - 1) / (C-1)
        out[(size_t)BATCH * CNUM + (size_t)b * CNUM + c] =
            (doti_s[c] * rc - 1.0f) * (1.0f / (float)(CNUM - 1));
    }

    // summary_avg_score[b] = mean_c (cand·summary) / (sn * cn)
    {
        float part = 0.f;
        if (tid < CNUM) part = dots_s[tid] * rcn_s[tid];
        #pragma unroll
        for (int off = 16; off > 0; off >>= 1) part += __shfl_xor(part, off, 32);
        if (lane == 0) red[0][w] = part;
    }
    __syncthreads();
    if (tid == 0) {
        float t = 0.f;
        #pragma unroll
        for (int i = 0; i < 8; ++i) t += red[0][i];
        // summary_score[b] = (doc·summary) / (sn * dn)
        out[2 * (size_t)BATCH * CNUM + b]         = scal[2] * rsn * rdn;
        out[2 * (size_t)BATCH * CNUM + BATCH + b] = t * rsn * (1.0f / (float)CNUM);
    }
}

extern "C" void kernel_launch(void* const* d_in, const int* in_sizes, int n_in,
                              void* d_out, int out_size, void* d_ws, size_t ws_size,
                              hipStream_t stream) {
    const float* doc  = (const float*)d_in[0];  // [256, 1024]
    const float* summ = (const float*)d_in[1];  // [256, 1024]
    const float* cand = (const float*)d_in[2];  // [256, 128, 1024]
    float* out = (float*)d_out;                 // 32768 + 32768 + 256 + 256
    rerank_wmma_kernel<<<BATCH, 256, 0, stream>>>(doc, summ, cand, out);
}